// HET_RelationalAttLayer_25056839205669
// MI455X (gfx1250) — compile-verified
//
#include <hip/hip_runtime.h>
#include <math.h>

typedef float v2f __attribute__((ext_vector_type(2)));
typedef float v8f __attribute__((ext_vector_type(8)));

#define N_NODES 100000
#define N_RELS 8
#define N_HEADS 4
#define IN_FEAT 64
#define OUT_FEAT 64
#define D_HEAD 16
#define LEAKY_SLOPE 0.2f

// Native f32 atomics (GLOBAL_ATOMIC_ADD_F32 / GLOBAL_ATOMIC_MAX_NUM_F32),
// avoiding CAS retry loops.
__device__ __forceinline__ void atomAddF(float* p, float v) { unsafeAtomicAdd(p, v); }
__device__ __forceinline__ void atomMaxF(float* p, float v) { unsafeAtomicMax(p, v); }

// ---------------- K0: init out=bias, m=-inf, denom=0 ----------------
__global__ void k0_init(float* __restrict__ out, const float* __restrict__ bias,
                        float* __restrict__ m, float* __restrict__ denom) {
    int i = blockIdx.x * blockDim.x + threadIdx.x;
    if (i < N_NODES * OUT_FEAT) out[i] = bias[i & 63];
    if (i < N_NODES * N_HEADS) { m[i] = -INFINITY; denom[i] = 0.0f; }
}

// ---------------- K1: per-relation node transform via f32 WMMA ----------------
// One wave per (relation, 16-node tile). C[16x64] = X[16x64] * W_r[64x64].
__global__ __launch_bounds__(32) void k1_transform(
    const float* __restrict__ x, const float* __restrict__ w,
    const float* __restrict__ attn_l, const float* __restrict__ attn_r,
    float* __restrict__ feat_all, float* __restrict__ el_all, float* __restrict__ er_all) {
    const int NTILES = N_NODES / 16;            // 6250, exact
    int tile = blockIdx.x;
    int r = tile / NTILES;
    int t = tile % NTILES;
    int lane = threadIdx.x & 31;
    int col  = lane & 15;                       // N / D index, and A-row index
    int khalf = (lane >> 4) * 2;                // K sub-offset within a k-step
    int base = t * 16;

    v8f c[4] = {v8f{}, v8f{}, v8f{}, v8f{}};

    #pragma unroll
    for (int k0 = 0; k0 < IN_FEAT; k0 += 4) {
        // A fragment: 16x4 of X. lane holds row (lane&15), K = k0+khalf+{0,1}
        const float* xp = x + (base + col) * IN_FEAT + k0 + khalf;
        v2f a; a.x = xp[0]; a.y = xp[1];
        #pragma unroll
        for (int nt = 0; nt < 4; ++nt) {
            // B fragment: 4x16 of W_r. B[k][n] = w[((r*4+nt)*64 + k)*16 + (n&15)]
            const float* wp = w + ((r * N_HEADS + nt) * IN_FEAT + (k0 + khalf)) * D_HEAD + col;
            v2f b; b.x = wp[0]; b.y = wp[D_HEAD];   // k and k+1 rows
            c[nt] = __builtin_amdgcn_wmma_f32_16x16x4_f32(
                false, a, false, b, (short)0, c[nt], false, false);
        }
    }

    // Store feat_all[r][row][64]; C VGPR g holds row g (lanes 0-15) / g+8 (lanes 16-31)
    int halfrow = (lane >> 4) * 8;
    #pragma unroll
    for (int nt = 0; nt < 4; ++nt) {
        #pragma unroll
        for (int g = 0; g < 8; ++g) {
            int row = base + g + halfrow;
            feat_all[((size_t)r * N_NODES + row) * OUT_FEAT + nt * 16 + col] = c[nt][g];
        }
    }

    // el/er: head h == n-tile h (D_HEAD==16). Reduce across 16-lane half-group.
    #pragma unroll
    for (int h = 0; h < 4; ++h) {
        float al = attn_l[(r * N_HEADS + h) * D_HEAD + col];
        float ar = attn_r[(r * N_HEADS + h) * D_HEAD + col];
        #pragma unroll
        for (int g = 0; g < 8; ++g) {
            float vl = c[h][g] * al;
            float vr = c[h][g] * ar;
            #pragma unroll
            for (int msk = 8; msk > 0; msk >>= 1) {
                vl += __shfl_xor(vl, msk, 16);
                vr += __shfl_xor(vr, msk, 16);
            }
            if (col == 0) {
                int row = base + g + halfrow;
                el_all[((size_t)r * N_NODES + row) * N_HEADS + h] = vl;
                er_all[((size_t)r * N_NODES + row) * N_HEADS + h] = vr;
            }
        }
    }
}

// ---------------- K2: edge logits + segment max ----------------
__global__ void k2_edge_logits(const int* __restrict__ src, const int* __restrict__ dst,
                               const int* __restrict__ rel,
                               const float* __restrict__ el_all, const float* __restrict__ er_all,
                               float* __restrict__ s_edge, float* __restrict__ m, int nE) {
    int e = blockIdx.x * blockDim.x + threadIdx.x;
    if (e >= nE) return;
    int rr = rel[e], ss = src[e], dd = dst[e];
    const float* elp = el_all + ((size_t)rr * N_NODES + ss) * N_HEADS;
    const float* erp = er_all + ((size_t)rr * N_NODES + dd) * N_HEADS;
    #pragma unroll
    for (int h = 0; h < N_HEADS; ++h) {
        float s = elp[h] + erp[h];
        s = (s > 0.0f) ? s : LEAKY_SLOPE * s;
        s_edge[(size_t)e * N_HEADS + h] = s;
        atomMaxF(&m[dd * N_HEADS + h], s);      // GLOBAL_ATOMIC_MAX_NUM_F32
    }
}

// ---------------- K3: exp + segment sum ----------------
__global__ void k3_edge_exp(const int* __restrict__ dst, float* __restrict__ s_edge,
                            const float* __restrict__ m, float* __restrict__ denom, int nE) {
    int e = blockIdx.x * blockDim.x + threadIdx.x;
    if (e >= nE) return;
    int dd = dst[e];
    #pragma unroll
    for (int h = 0; h < N_HEADS; ++h) {
        float ev = __expf(s_edge[(size_t)e * N_HEADS + h] - m[dd * N_HEADS + h]);
        s_edge[(size_t)e * N_HEADS + h] = ev;
        atomAddF(&denom[dd * N_HEADS + h], ev); // GLOBAL_ATOMIC_ADD_F32
    }
}

// ---------------- K4: weighted aggregation ----------------
__global__ void k4_aggregate(const int* __restrict__ src, const int* __restrict__ dst,
                             const int* __restrict__ rel,
                             const float* __restrict__ s_edge, const float* __restrict__ denom,
                             const float* __restrict__ feat_all, float* __restrict__ out, int nE) {
    int idx = blockIdx.x * blockDim.x + threadIdx.x;
    if (idx >= nE * N_HEADS) return;
    int e = idx >> 2;
    int h = idx & 3;
    int rr = rel[e], ss = src[e], dd = dst[e];
    float d = denom[dd * N_HEADS + h];
    float wgt = s_edge[(size_t)e * N_HEADS + h] / fmaxf(d, 1e-16f);
    const float4* fp = (const float4*)(feat_all + ((size_t)rr * N_NODES + ss) * OUT_FEAT + h * D_HEAD);
    float* op = out + (size_t)dd * OUT_FEAT + h * D_HEAD;
    #pragma unroll
    for (int q = 0; q < 4; ++q) {
        float4 f = fp[q];
        atomAddF(&op[q * 4 + 0], f.x * wgt);
        atomAddF(&op[q * 4 + 1], f.y * wgt);
        atomAddF(&op[q * 4 + 2], f.z * wgt);
        atomAddF(&op[q * 4 + 3], f.w * wgt);
    }
}

extern "C" void kernel_launch(void* const* d_in, const int* in_sizes, int n_in,
                              void* d_out, int out_size, void* d_ws, size_t ws_size,
                              hipStream_t stream) {
    const float* x      = (const float*)d_in[0];
    const int*   src    = (const int*)d_in[1];
    const int*   dst    = (const int*)d_in[2];
    const int*   rel    = (const int*)d_in[3];
    const float* w      = (const float*)d_in[4];
    const float* attn_l = (const float*)d_in[5];
    const float* attn_r = (const float*)d_in[6];
    const float* bias   = (const float*)d_in[7];
    float* out = (float*)d_out;
    const int nE = in_sizes[1];

    // workspace layout
    char* ws = (char*)d_ws;
    const size_t FEAT_B = (size_t)N_RELS * N_NODES * OUT_FEAT * 4;   // 204.8 MB
    const size_t ELR_B  = (size_t)N_RELS * N_NODES * N_HEADS * 4;    // 12.8 MB
    const size_t MD_B   = (size_t)N_NODES * N_HEADS * 4;             // 1.6 MB
    float* feat_all = (float*)ws;
    float* el_all   = (float*)(ws + FEAT_B);
    float* er_all   = (float*)(ws + FEAT_B + ELR_B);
    float* mbuf     = (float*)(ws + FEAT_B + 2 * ELR_B);
    float* denom    = (float*)(ws + FEAT_B + 2 * ELR_B + MD_B);
    float* s_edge   = (float*)(ws + FEAT_B + 2 * ELR_B + 2 * MD_B);

    // K0: init
    {
        int total = N_NODES * OUT_FEAT;
        k0_init<<<(total + 255) / 256, 256, 0, stream>>>(out, bias, mbuf, denom);
    }
    // K1: WMMA transform, one wave per (relation, 16-node tile)
    {
        int tiles = N_RELS * (N_NODES / 16);
        k1_transform<<<tiles, 32, 0, stream>>>(x, w, attn_l, attn_r, feat_all, el_all, er_all);
    }
    // K2: logits + max
    k2_edge_logits<<<(nE + 255) / 256, 256, 0, stream>>>(src, dst, rel, el_all, er_all, s_edge, mbuf, nE);
    // K3: exp + denom
    k3_edge_exp<<<(nE + 255) / 256, 256, 0, stream>>>(dst, s_edge, mbuf, denom, nE);
    // K4: aggregate
    k4_aggregate<<<(nE * N_HEADS + 255) / 256, 256, 0, stream>>>(src, dst, rel, s_edge, denom, feat_all, out, nE);
}